// SpatialAttention_86114094285446
// MI455X (gfx1250) — compile-verified
//
#include <hip/hip_runtime.h>
#include <hip/hip_bf16.h>

// MI455X (gfx1250, wave32) implementation.
// Pass 1-3: pooling, Q/K projection, softmax attention (tiny, VALU).
// Pass 4:   y = A(64x64) @ V(64x16384) per (b,h) via v_wmma_f32_16x16x4_f32, y -> d_out.
// Pass 5:   out = [Wc_o|Wc_x](256x512) @ [y;x](512x512) per (b,s); y slice staged into
//           LDS with async global->LDS (ASYNCcnt) copies, then the same d_out footprint
//           is overwritten in place with the conv result.

typedef __attribute__((ext_vector_type(2))) float v2f;
typedef __attribute__((ext_vector_type(8))) float v8f;

#define DIM 256
#define T 512
#define S 64
#define B 8
#define H 8
#define ROW (DIM * T)        // 131072 floats per (b,s) slice
#define HSTRIDE (32 * T)     // 16384 floats per head inside a slice

static __device__ __forceinline__ v8f wmma_f32(v2f a, v2f b, v8f c) {
  return __builtin_amdgcn_wmma_f32_16x16x4_f32(false, a, false, b, (short)0, c,
                                               false, false);
}

// ---------- Pass 1: pooled[bs,c] = mean_t x[bs,c,t]; one wave per row ----------
__global__ void k_pool(const float* __restrict__ x, float* __restrict__ pooled) {
  int row  = blockIdx.x * 8 + (threadIdx.x >> 5);   // 0 .. 512*256-1
  int lane = threadIdx.x & 31;
  const float* p = x + (size_t)row * T;
  float s = 0.f;
  #pragma unroll
  for (int i = 0; i < T / 32; ++i) s += p[lane + i * 32];
  #pragma unroll
  for (int off = 16; off > 0; off >>= 1) s += __shfl_down(s, off, 32);
  if (lane == 0) pooled[row] = s * (1.0f / (float)T);
}

// ---------- Pass 2: q/k = pooled @ Wq^T / Wk^T ----------
__global__ void k_qk(const float* __restrict__ pooled,
                     const float* __restrict__ Wq, const float* __restrict__ Wk,
                     float* __restrict__ q, float* __restrict__ k) {
  int idx   = blockIdx.x * blockDim.x + threadIdx.x;  // 0 .. 2*131072-1
  int which = idx >> 17;                              // 0 -> q, 1 -> k
  int t     = idx & (131072 - 1);
  int bs = t >> 8, o = t & 255;
  const float* w  = (which ? Wk : Wq) + (size_t)o * DIM;
  const float* pr = pooled + (size_t)bs * DIM;
  float acc = 0.f;
  #pragma unroll 8
  for (int c = 0; c < DIM; ++c) acc += pr[c] * w[c];
  (which ? k : q)[t] = acc;
}

// ---------- Pass 3: scores + softmax -> a[b,h,s,k]; one wave per (b,h,s) ----------
__global__ void k_attn(const float* __restrict__ q, const float* __restrict__ k,
                       const float* __restrict__ mask, float* __restrict__ a) {
  int row  = blockIdx.x * 8 + (threadIdx.x >> 5);  // b*512 + h*64 + s
  int lane = threadIdx.x & 31;
  int b = row >> 9, h = (row >> 6) & 7, s = row & 63;
  const float* qv = q + ((size_t)(b * S + s) * 256 + h * 32);
  float sc[2];
  #pragma unroll
  for (int j = 0; j < 2; ++j) {
    int kk = lane + 32 * j;
    const float* kv = k + ((size_t)(b * S + kk) * 256 + h * 32);
    float d = 0.f;
    #pragma unroll
    for (int c = 0; c < 32; ++c) d += qv[c] * kv[c];
    sc[j] = d * 0.17677669529663687f + mask[b * S + kk];  // 1/sqrt(32)
  }
  float m = fmaxf(sc[0], sc[1]);
  #pragma unroll
  for (int off = 16; off > 0; off >>= 1) m = fmaxf(m, __shfl_xor(m, off, 32));
  float e0 = __expf(sc[0] - m), e1 = __expf(sc[1] - m);
  float sum = e0 + e1;
  #pragma unroll
  for (int off = 16; off > 0; off >>= 1) sum += __shfl_xor(sum, off, 32);
  float inv = 1.0f / sum;
  a[(size_t)row * 64 + lane]      = e0 * inv;
  a[(size_t)row * 64 + lane + 32] = e1 * inv;
}

// ---------- Pass 4: per (b,h): Y(64x16384) = A(64x64) @ V(64x16384), Y -> d_out ----------
// grid: (ntile=256, bh=64), block 128 (4 waves); wave w owns M-tile w (s rows w*16..+16),
// and N columns [n0, n0+64) as four 16-wide WMMA subtiles. K = 64 attention keys.
__global__ void k_ypass(const float* __restrict__ x, const float* __restrict__ a,
                        float* __restrict__ y) {
  int bh = blockIdx.y;
  int b = bh >> 3, h = bh & 7;
  int mt   = threadIdx.x >> 5;
  int lane = threadIdx.x & 31;
  int lm = lane & 15, hi = lane >> 4;
  int n0 = blockIdx.x * 64;

  const float* arow  = a + ((size_t)bh * S + (mt * 16 + lm)) * S;        // a[b,h,m,:]
  const float* vbase = x + (size_t)(b * S) * ROW + (size_t)h * HSTRIDE;  // + k*ROW + n

  v8f acc[4] = {};
  for (int k0 = 0; k0 < S; k0 += 4) {
    int kb = k0 + 2 * hi;
    v2f A = *(const v2f*)(arow + kb);                     // A[m, kb], A[m, kb+1]
    const float* v0 = vbase + (size_t)kb * ROW + n0 + lm; // V[kb, n]
    const float* v1 = v0 + ROW;                           // V[kb+1, n]
    #pragma unroll
    for (int j = 0; j < 4; ++j) {
      v2f Bv;
      Bv.x = v0[j * 16];
      Bv.y = v1[j * 16];
      acc[j] = wmma_f32(A, Bv, acc[j]);
    }
  }
  #pragma unroll
  for (int j = 0; j < 4; ++j) {
    int n = n0 + j * 16 + lm;
    #pragma unroll
    for (int r = 0; r < 8; ++r) {
      int sr = mt * 16 + r + 8 * hi;  // M = r (lanes 0-15) / r+8 (lanes 16-31)
      y[(size_t)(b * S + sr) * ROW + (size_t)h * HSTRIDE + n] = acc[j][r];
    }
  }
}

// ---------- Pass 5: per (bs, t-tile of 64): out(256x64) = Wc(256x512) @ [y;x](512x64) ----------
// y slice (64KB) staged to LDS via CDNA5 async global->LDS, then overwritten in place.
// block 256 (8 waves); wave w owns output rows [w*32, w*32+32) as 2 M-tiles x 4 N-tiles.
__global__ void k_outpass(const float* __restrict__ x, const float* __restrict__ wc,
                          float* __restrict__ out) {
  __shared__ float yl[DIM * 64];
  int bs = blockIdx.y;
  int t0 = blockIdx.x * 64;
  size_t base = (size_t)bs * ROW + t0;  // + c*T + tt addresses both y-in and out

  // Stage y[bs, :, t0:t0+64] into LDS. 16B chunks, async engine, no VGPR round-trip.
  {
    uint32_t lds_base = (uint32_t)(uintptr_t)(void*)yl;  // low 32 bits = LDS offset
    for (int i = threadIdx.x; i < (DIM * 64) / 4; i += 256) {
      int c = i >> 4, tt = (i & 15) * 4;                 // 16 x b128 per row of 64
      const float* gsrc = out + base + (size_t)c * T + tt;
      uint32_t ldst = lds_base + (uint32_t)i * 16u;
      asm volatile("global_load_async_to_lds_b128 %0, %1, off"
                   :
                   : "v"(ldst), "v"(gsrc)
                   : "memory");
    }
    asm volatile("s_wait_asynccnt 0x0" ::: "memory");
  }
  __syncthreads();

  int wv   = threadIdx.x >> 5;
  int lane = threadIdx.x & 31;
  int lm = lane & 15, hi = lane >> 4;

  v8f acc[2][4] = {};
  const float* xbase = x + base;  // x[bs, c, t0 + n]

  for (int k0 = 0; k0 < 512; k0 += 4) {
    int kb = k0 + 2 * hi;
    v2f A0 = *(const v2f*)(wc + (size_t)(wv * 32 + lm) * 512 + kb);
    v2f A1 = *(const v2f*)(wc + (size_t)(wv * 32 + 16 + lm) * 512 + kb);
    #pragma unroll
    for (int j = 0; j < 4; ++j) {
      int n = j * 16 + lm;
      v2f Bv;
      if (k0 < 256) {  // uniform branch: cat rows 0..255 come from y (LDS)
        Bv.x = yl[kb * 64 + n];
        Bv.y = yl[(kb + 1) * 64 + n];
      } else {         // cat rows 256..511 come from x (global, L2-hot)
        Bv.x = xbase[(size_t)(kb - 256) * T + n];
        Bv.y = xbase[(size_t)(kb - 255) * T + n];
      }
      acc[0][j] = wmma_f32(A0, Bv, acc[0][j]);
      acc[1][j] = wmma_f32(A1, Bv, acc[1][j]);
    }
  }

  #pragma unroll
  for (int mi = 0; mi < 2; ++mi)
    #pragma unroll
    for (int j = 0; j < 4; ++j) {
      int n = j * 16 + lm;
      #pragma unroll
      for (int r = 0; r < 8; ++r) {
        int o = wv * 32 + mi * 16 + r + 8 * hi;
        out[base + (size_t)o * T + n] = acc[mi][j][r];
      }
    }
}

extern "C" void kernel_launch(void* const* d_in, const int* in_sizes, int n_in,
                              void* d_out, int out_size, void* d_ws, size_t ws_size,
                              hipStream_t stream) {
  const float* x    = (const float*)d_in[0];
  const float* Wq   = (const float*)d_in[1];
  const float* Wk   = (const float*)d_in[2];
  const float* Wc   = (const float*)d_in[3];
  const float* mask = (const float*)d_in[4];
  float* out = (float*)d_out;

  char* ws = (char*)d_ws;
  float* pooled = (float*)(ws);                       // 512*256 floats (512 KB)
  float* q      = (float*)(ws + (512u << 10));        // 512 KB
  float* k      = (float*)(ws + (1024u << 10));       // 512 KB
  float* a      = (float*)(ws + (1536u << 10));       // 8*8*64*64 floats (1 MB)

  k_pool<<<dim3(16384), dim3(256), 0, stream>>>(x, pooled);
  k_qk<<<dim3(1024), dim3(256), 0, stream>>>(pooled, Wq, Wk, q, k);
  k_attn<<<dim3(512), dim3(256), 0, stream>>>(q, k, mask, a);
  k_ypass<<<dim3(256, 64), dim3(128), 0, stream>>>(x, a, out);     // y -> d_out
  k_outpass<<<dim3(8, 512), dim3(256), 0, stream>>>(x, Wc, out);   // in-place conv
}